// DeepVCP_31129922961420
// MI455X (gfx1250) — compile-verified
//
#include <hip/hip_runtime.h>
#include <hip/hip_bf16.h>

#define B_      8
#define D_      6
#define N_      131072
#define KTOP    64
#define NPOINT  64
#define NSAMPLE 32

typedef __attribute__((ext_vector_type(2))) float v2f;
typedef __attribute__((ext_vector_type(8))) float v8f;

// ---------------------------------------------------------------------------
// Pass 0: fold W2/w_score/b2/b_score into v[64] and scalar c.
//   score[n] = v . relu(W1 x_n + b1) + c
// ---------------------------------------------------------------------------
__global__ __launch_bounds__(64) void precompute_v_kernel(
    const float* __restrict__ W2, const float* __restrict__ b2,
    const float* __restrict__ wsc, const float* __restrict__ bsc,
    float* __restrict__ vvec, float* __restrict__ cconst)
{
    const int o = threadIdx.x;
    float vo = 0.0f;
    for (int p = 0; p < 32; ++p) vo += wsc[p] * W2[p * 64 + o];
    vvec[o] = vo;
    if (o == 0) {
        float cs = bsc[0];
        for (int p = 0; p < 32; ++p) cs += wsc[p] * b2[p];
        cconst[0] = cs;
    }
}

// ---------------------------------------------------------------------------
// Pass 1: fused score kernel, one wave per 16-point tile.
// V_WMMA_F32_16X16X4_F32 for the 6->64 GEMM (D padded to 8 -> 2 k-steps).
// ---------------------------------------------------------------------------
__global__ __launch_bounds__(256) void score_kernel(
    const float* __restrict__ src, const float* __restrict__ W1,
    const float* __restrict__ b1, const float* __restrict__ vvec,
    const float* __restrict__ cconst, float* __restrict__ scores)
{
    const int lane = threadIdx.x & 31;
    const int wave = threadIdx.x >> 5;
    const int tile = blockIdx.x * 8 + wave;
    const int tilesPerB = N_ / 16;
    const int b  = tile / tilesPerB;
    const int n0 = (tile % tilesPerB) * 16;
    const int half = lane >> 4;      // which 16-lane half
    const int nn   = lane & 15;

    // ---- A operand: 16 points x K(=8 padded).  f32 16x4 A layout:
    // lanes 0-15 supply K={4s+0,4s+1}, lanes 16-31 K={4s+2,4s+3}.
    // Branch-free padding: clamp index, multiply by 0/1 mask.
    const float* sb = src + (size_t)b * D_ * N_;
    const int n = n0 + nn;
    v2f a[2];
#pragma unroll
    for (int s = 0; s < 2; ++s) {
        const int   kb = 4 * s + 2 * half;
        const float m0 = (kb     < D_) ? 1.0f : 0.0f;
        const float m1 = (kb + 1 < D_) ? 1.0f : 0.0f;
        const int   k0 = (kb     < D_) ? kb     : 0;
        const int   k1 = (kb + 1 < D_) ? kb + 1 : 0;
        a[s][0] = sb[(size_t)k0 * N_ + n] * m0;
        a[s][1] = sb[(size_t)k1 * N_ + n] * m1;
    }

    float partial[8];
#pragma unroll
    for (int r = 0; r < 8; ++r) partial[r] = 0.0f;

#pragma unroll
    for (int c = 0; c < 4; ++c) {        // 4 chunks of 16 hidden units
        const int o = 16 * c + nn;       // this lane's output column
        // B operand: K x 16 slice of W1^T (mirror layout of A)
        v2f bm[2];
#pragma unroll
        for (int s = 0; s < 2; ++s) {
            const int   kb = 4 * s + 2 * half;
            const float m0 = (kb     < D_) ? 1.0f : 0.0f;
            const float m1 = (kb + 1 < D_) ? 1.0f : 0.0f;
            const int   k0 = (kb     < D_) ? kb     : 0;
            const int   k1 = (kb + 1 < D_) ? kb + 1 : 0;
            bm[s][0] = W1[o * D_ + k0] * m0;
            bm[s][1] = W1[o * D_ + k1] * m1;
        }
        const float bias = b1[o];
        v8f acc;
#pragma unroll
        for (int r = 0; r < 8; ++r) acc[r] = bias;
        acc = __builtin_amdgcn_wmma_f32_16x16x4_f32(false, a[0], false, bm[0],
                                                    (short)0, acc, false, false);
        acc = __builtin_amdgcn_wmma_f32_16x16x4_f32(false, a[1], false, bm[1],
                                                    (short)0, acc, false, false);
        const float vo = vvec[o];
#pragma unroll
        for (int r = 0; r < 8; ++r) {
            const float hv = acc[r] > 0.0f ? acc[r] : 0.0f;   // relu
            partial[r] += hv * vo;
        }
    }

    // Butterfly-reduce across the 16 lanes of each half (sum over 64 columns)
#pragma unroll
    for (int off = 1; off <= 8; off <<= 1) {
#pragma unroll
        for (int r = 0; r < 8; ++r)
            partial[r] += __shfl_xor(partial[r], off, 32);
    }

    if (nn == 0) {
        const float cs = cconst[0];
        // C/D layout: VGPR r holds row M = r (lanes 0-15) or 8+r (lanes 16-31)
#pragma unroll
        for (int r = 0; r < 8; ++r) {
            const int m = 8 * half + r;
            scores[(size_t)b * N_ + n0 + m] = partial[r] + cs;
        }
    }
}

// ---------------------------------------------------------------------------
// Pass 2: per-batch exact top-64 (64 x deterministic block argmax,
// tie-break = lowest index, matching lax.top_k stability).  1024 threads.
// ---------------------------------------------------------------------------
__global__ __launch_bounds__(1024) void topk_kernel(float* __restrict__ scores,
                                                    int* __restrict__ keyidx)
{
    const int b = blockIdx.x;
    const int tid = threadIdx.x;
    __shared__ float sv[1024];
    __shared__ int   si[1024];
    float* sc = scores + (size_t)b * N_;

    for (int it = 0; it < KTOP; ++it) {
        float best = -__builtin_inff();
        int   bi   = N_;
        for (int j = tid; j < N_; j += 1024) {
            const float s = sc[j];
            if (s > best) { best = s; bi = j; }   // ascending j -> lowest idx
        }
        sv[tid] = best; si[tid] = bi;
        __syncthreads();
        for (int off = 512; off > 0; off >>= 1) {
            if (tid < off) {
                const float ov = sv[tid + off];
                const int   oi = si[tid + off];
                if (ov > sv[tid] || (ov == sv[tid] && oi < si[tid])) {
                    sv[tid] = ov; si[tid] = oi;
                }
            }
            __syncthreads();
        }
        if (tid == 0) {
            keyidx[b * KTOP + it] = si[0];
            sc[si[0]] = -__builtin_inff();
        }
        __syncthreads();
    }
}

// ---------------------------------------------------------------------------
// Gather keypoints -> output region 0 and xyz scratch
// ---------------------------------------------------------------------------
__global__ void gather_keypts_kernel(const float* __restrict__ src,
                                     const int* __restrict__ keyidx,
                                     float* __restrict__ out,
                                     float* __restrict__ xyz)
{
    const int t = blockIdx.x * blockDim.x + threadIdx.x;
    if (t >= B_ * KTOP * D_) return;
    const int d = t % D_;
    const int i = (t / D_) % KTOP;
    const int b = t / (D_ * KTOP);
    const int idx = keyidx[b * KTOP + i];
    const float v = src[((size_t)b * D_ + d) * N_ + idx];
    out[(b * KTOP + i) * D_ + d] = v;
    if (d < 3) xyz[(b * KTOP + i) * 3 + d] = v;
}

// ---------------------------------------------------------------------------
// Farthest point sampling, exactly mirroring the reference loop semantics.
// ---------------------------------------------------------------------------
__global__ __launch_bounds__(64) void fps_kernel(const float* __restrict__ xyz,
                                                 int* __restrict__ fpsidx,
                                                 float* __restrict__ newxyz)
{
    const int b = blockIdx.x, tid = threadIdx.x;
    __shared__ float px[64], py[64], pz[64], dist[64];
    __shared__ int far;
    const float* xb = xyz + b * KTOP * 3;
    px[tid] = xb[tid * 3 + 0];
    py[tid] = xb[tid * 3 + 1];
    pz[tid] = xb[tid * 3 + 2];
    dist[tid] = 1e10f;
    if (tid == 0) far = 0;
    __syncthreads();
    for (int i = 0; i < NPOINT; ++i) {
        const int f = far;
        if (tid == 0) {
            fpsidx[b * NPOINT + i] = f;
            newxyz[(b * NPOINT + i) * 3 + 0] = px[f];
            newxyz[(b * NPOINT + i) * 3 + 1] = py[f];
            newxyz[(b * NPOINT + i) * 3 + 2] = pz[f];
        }
        const float dx = px[tid] - px[f], dy = py[tid] - py[f], dz = pz[tid] - pz[f];
        const float d = dx * dx + dy * dy + dz * dz;
        if (d < dist[tid]) dist[tid] = d;
        __syncthreads();
        if (tid == 0) {           // first-occurrence argmax (jnp.argmax)
            float bd = dist[0]; int bj = 0;
            for (int j = 1; j < 64; ++j)
                if (dist[j] > bd) { bd = dist[j]; bj = j; }
            far = bj;
        }
        __syncthreads();
    }
}

// ---------------------------------------------------------------------------
// Ball query: sorted in-radius indices == ascending scan; pad with first.
// ---------------------------------------------------------------------------
__global__ void ball_kernel(const float* __restrict__ xyz,
                            const float* __restrict__ newxyz,
                            int* __restrict__ gidx)
{
    const int t = blockIdx.x * blockDim.x + threadIdx.x;
    if (t >= B_ * NPOINT) return;
    const int b = t / NPOINT;
    const float cx = newxyz[t * 3 + 0];
    const float cy = newxyz[t * 3 + 1];
    const float cz = newxyz[t * 3 + 2];
    const float* xb = xyz + b * KTOP * 3;
    int cnt = 0, first = -1;
    int* g = gidx + t * NSAMPLE;
    for (int j = 0; j < KTOP; ++j) {
        const float dx = xb[j * 3 + 0] - cx;
        const float dy = xb[j * 3 + 1] - cy;
        const float dz = xb[j * 3 + 2] - cz;
        const float d2 = dx * dx + dy * dy + dz * dz;
        if (!(d2 > 1.0f)) {                       // radius^2 = 1.0
            if (first < 0) first = j;
            if (cnt < NSAMPLE) g[cnt] = j;
            cnt++;
        }
    }
    if (first < 0) first = KTOP - 1;              // unreachable (self dist 0)
    for (int k = cnt; k < NSAMPLE; ++k) g[k] = first;
}

// ---------------------------------------------------------------------------
// Recompute full 32-dim deep_feat only for the 512 selected keypoints.
// ---------------------------------------------------------------------------
__global__ void feat_kernel(const float* __restrict__ src,
                            const int* __restrict__ keyidx,
                            const float* __restrict__ W1,
                            const float* __restrict__ b1,
                            const float* __restrict__ W2,
                            const float* __restrict__ b2,
                            float* __restrict__ featsel)
{
    const int t = blockIdx.x * blockDim.x + threadIdx.x;
    if (t >= B_ * KTOP) return;
    const int b = t / KTOP;
    const int idx = keyidx[t];
    float x[D_];
    for (int d = 0; d < D_; ++d) x[d] = src[((size_t)b * D_ + d) * N_ + idx];
    float f[32];
    for (int p = 0; p < 32; ++p) f[p] = b2[p];
    for (int o = 0; o < 64; ++o) {
        float h = b1[o];
        for (int d = 0; d < D_; ++d) h += W1[o * D_ + d] * x[d];
        h = h > 0.0f ? h : 0.0f;
        for (int p = 0; p < 32; ++p) f[p] += W2[p * 64 + o] * h;
    }
    for (int p = 0; p < 32; ++p) featsel[t * 32 + p] = f[p];
}

// ---------------------------------------------------------------------------
// Assemble src_keyfeats_cat = [grouped_xyz(3) | deep_feat(32)] per (b,p,s).
// ---------------------------------------------------------------------------
__global__ void assemble_kernel(const float* __restrict__ xyz,
                                const float* __restrict__ newxyz,
                                const int* __restrict__ gidx,
                                const float* __restrict__ featsel,
                                float* __restrict__ out)
{
    const int t = blockIdx.x * blockDim.x + threadIdx.x;
    if (t >= B_ * NPOINT * NSAMPLE) return;
    const int bp = t / NSAMPLE;        // b*NPOINT + p
    const int b  = bp / NPOINT;
    const int g  = gidx[t];
    float* o = out + B_ * KTOP * D_ + (size_t)t * 35;
    const float* xb = xyz + (b * KTOP + g) * 3;
    o[0] = xb[0] - newxyz[bp * 3 + 0];
    o[1] = xb[1] - newxyz[bp * 3 + 1];
    o[2] = xb[2] - newxyz[bp * 3 + 2];
    const float* f = featsel + (b * KTOP + g) * 32;
    for (int k = 0; k < 32; ++k) o[3 + k] = f[k];
}

extern "C" void kernel_launch(void* const* d_in, const int* in_sizes, int n_in,
                              void* d_out, int out_size, void* d_ws, size_t ws_size,
                              hipStream_t stream)
{
    const float* src = (const float*)d_in[0];   // src_pts (B,D,N)
    // d_in[1] = tgt_pts (unused by reference)
    const float* W1  = (const float*)d_in[2];
    const float* b1v = (const float*)d_in[3];
    const float* W2  = (const float*)d_in[4];
    const float* b2v = (const float*)d_in[5];
    const float* wsc = (const float*)d_in[6];
    const float* bsc = (const float*)d_in[7];
    float* out = (float*)d_out;

    // Workspace layout (~4.2 MB total)
    char* ws = (char*)d_ws;
    size_t off = 0;
    float* scores  = (float*)(ws + off); off += (size_t)B_ * N_ * sizeof(float);
    int*   keyidx  = (int*)  (ws + off); off += (size_t)B_ * KTOP * sizeof(int);
    float* xyz     = (float*)(ws + off); off += (size_t)B_ * KTOP * 3 * sizeof(float);
    int*   fpsidx  = (int*)  (ws + off); off += (size_t)B_ * NPOINT * sizeof(int);
    float* newxyz  = (float*)(ws + off); off += (size_t)B_ * NPOINT * 3 * sizeof(float);
    int*   gidx    = (int*)  (ws + off); off += (size_t)B_ * NPOINT * NSAMPLE * sizeof(int);
    float* featsel = (float*)(ws + off); off += (size_t)B_ * KTOP * 32 * sizeof(float);
    float* vvec    = (float*)(ws + off); off += 64 * sizeof(float);
    float* cconst  = (float*)(ws + off);

    precompute_v_kernel<<<1, 64, 0, stream>>>(W2, b2v, wsc, bsc, vvec, cconst);
    score_kernel<<<(B_ * (N_ / 16)) / 8, 256, 0, stream>>>(src, W1, b1v,
                                                           vvec, cconst, scores);
    topk_kernel<<<B_, 1024, 0, stream>>>(scores, keyidx);
    gather_keypts_kernel<<<(B_ * KTOP * D_ + 255) / 256, 256, 0, stream>>>(
        src, keyidx, out, xyz);
    fps_kernel<<<B_, 64, 0, stream>>>(xyz, fpsidx, newxyz);
    ball_kernel<<<(B_ * NPOINT + 255) / 256, 256, 0, stream>>>(xyz, newxyz, gidx);
    feat_kernel<<<(B_ * KTOP + 255) / 256, 256, 0, stream>>>(src, keyidx, W1, b1v,
                                                             W2, b2v, featsel);
    assemble_kernel<<<(B_ * NPOINT * NSAMPLE + 255) / 256, 256, 0, stream>>>(
        xyz, newxyz, gidx, featsel, out);
}